// PManifoldLayer_4449586119171
// MI455X (gfx1250) — compile-verified
//
#include <hip/hip_runtime.h>
#include <math.h>

typedef __attribute__((ext_vector_type(2))) float v2f;
typedef __attribute__((ext_vector_type(8))) float v8f;

#define KDIM   32
#define MDIM   16
#define ROWSZ  (MDIM + 1)     // 17 floats per dgm row
#define NPTS   2048
#define BATCH  32
#define NWAVES 8
#define NCHUNK (NPTS / 16)    // 128 chunks of 16 rows
#define EPSF   1e-12f

__device__ __forceinline__ v8f wmma4(v2f a, v2f b, v8f c) {
    // D = A(16x4,f32) * B(4x16,f32) + C(16x16,f32)
    // args: (neg_a, A, neg_b, B, c_mod, C, reuse_a, reuse_b)
    return __builtin_amdgcn_wmma_f32_16x16x4_f32(false, a, false, b, (short)0, c,
                                                 false, false);
}

// LDS byte offset of a __shared__ object (DS-instruction address)
__device__ __forceinline__ unsigned lds_offset(const void* p) {
    return (unsigned)(uintptr_t)(__attribute__((address_space(3))) const char*)p;
}

// Stage one 16-row chunk (272 contiguous floats = 68 x 16B, 16B-aligned since the
// chunk stride is 1088B) into LDS via async DMA. Exactly 3 async instructions per
// wave per chunk (the guarded one always issues: lanes 0-3 are always active).
__device__ __forceinline__ void stage_chunk_async(unsigned lbase, const float* src,
                                                  int lane) {
    const unsigned long long gbase = (unsigned long long)(uintptr_t)src;
    unsigned           l0 = lbase + (unsigned)lane * 16u;
    unsigned long long g0 = gbase + (unsigned long long)lane * 16ull;
    asm volatile("global_load_async_to_lds_b128 %0, %1, off"
                 :: "v"(l0), "v"(g0) : "memory");
    asm volatile("global_load_async_to_lds_b128 %0, %1, off"
                 :: "v"(l0 + 512u), "v"(g0 + 512ull) : "memory");
    if (lane < 4) {
        unsigned           l2 = lbase + (64u + (unsigned)lane) * 16u;
        unsigned long long g2 = gbase + (64ull + (unsigned)lane) * 16ull;
        asm volatile("global_load_async_to_lds_b128 %0, %1, off"
                     :: "v"(l2), "v"(g2) : "memory");
    }
}

__global__ __launch_bounds__(256) void pmanifold_kernel(
    const float* __restrict__ dgm,      // (B, N, 17)
    const float* __restrict__ theta,    // (K, M) = (32, 16)
    const float* __restrict__ class_w,  // (2,)
    float* __restrict__ out)            // (B, K*M)
{
    __shared__ __align__(16) float chunkLds[NWAVES][2][16 * ROWSZ];  // double-buffered
    __shared__ __align__(16) float aLds[NWAVES][16][KDIM];           // a[n,k] scratch
    __shared__ float S_lds[KDIM * MDIM];            // block-reduced S
    __shared__ float Scum[KDIM * MDIM];             // cumsum over k
    __shared__ int   s_n0;                          // first invalid index

    const int b    = blockIdx.x;
    const int tid  = threadIdx.x;
    const int wave = tid >> 5;
    const int lane = tid & 31;
    const int hl   = lane >> 4;   // 0: lanes 0-15, 1: lanes 16-31
    const int l16  = lane & 15;

    const float* dgm_b = dgm + (size_t)b * NPTS * ROWSZ;

    if (tid == 0) s_n0 = NPTS;
    for (int i = tid; i < KDIM * MDIM; i += blockDim.x) S_lds[i] = 0.f;
    __syncthreads();

    // ---- Phase 0: cumprod(valid_pt) == (n < first invalid index) ----
    for (int n = tid; n < NPTS; n += blockDim.x) {
        const float* row = dgm_b + (size_t)n * ROWSZ;
        bool nz = false;
        #pragma unroll
        for (int j = 0; j < ROWSZ; ++j) nz |= (row[j] != 0.f);
        int hom = (int)row[0];
        if (!((hom <= 1) && nz)) atomicMin(&s_n0, n);
    }
    __syncthreads();
    const int   n0  = s_n0;
    const float cw0 = class_w[0];
    const float cw1 = class_w[1];

    // ---- Loop-invariant B fragments for GEMM1: theta^2 transposed (4m x 16k) ----
    v2f thB[4][2];
    #pragma unroll
    for (int s = 0; s < 4; ++s) {
        #pragma unroll
        for (int t = 0; t < 2; ++t) {
            int k  = l16 + 16 * t;
            int m0 = 4 * s + 2 * hl;
            float t0 = theta[k * MDIM + m0];
            float t1 = theta[k * MDIM + m0 + 1];
            v2f v = {t0 * t0, t1 * t1};
            thB[s][t] = v;
        }
    }

    v8f acc0 = {};  // Spart tile k=0..15  (rows k, cols m)
    v8f acc1 = {};  // Spart tile k=16..31

    const unsigned lb[2] = { lds_offset(chunkLds[wave][0]),
                             lds_offset(chunkLds[wave][1]) };

    // Prologue: kick off DMA for this wave's first chunk.
    stage_chunk_async(lb[0], dgm_b + (size_t)wave * 16 * ROWSZ, lane);

    int buf = 0;
    // ---- Main loop: each wave owns 16 chunks (uniform -> EXEC all-ones at WMMA) ----
    for (int c = wave; c < NCHUNK; c += NWAVES) {
        const int n_base = c * 16;
        const int nx     = c + NWAVES;
        if (nx < NCHUNK) {
            const float* nsrc = dgm_b + (size_t)nx * 16 * ROWSZ;
            __builtin_prefetch(nsrc + 16 * ROWSZ * NWAVES, 0, 1);  // warm L2 (c+2)
            // Close WAR window: prior DS reads of buf^1 must be complete before
            // async writes can land there (DS pipe vs async writes are unordered).
            asm volatile("s_wait_dscnt 0" ::: "memory");
            stage_chunk_async(lb[buf ^ 1], nsrc, lane);
            // 3 newest async ops (next chunk) may stay in flight; the 3 older
            // ones (current chunk) are done: async loads complete in order.
            asm volatile("s_wait_asynccnt 3" ::: "memory");
        } else {
            asm volatile("s_wait_asynccnt 0" ::: "memory");
        }
        const float* ch = chunkLds[wave][buf];

        // ---- GEMM1: su(16n x 32k) = Y^2 (16n x 16m) * Theta^2T (16m x 32k) ----
        v8f su0 = {}, su1 = {};
        #pragma unroll
        for (int s = 0; s < 4; ++s) {
            // A frag: row n = l16, contraction cols m = 4s+2*hl, +1
            float y0 = ch[l16 * ROWSZ + 1 + 4 * s + 2 * hl];
            float y1 = ch[l16 * ROWSZ + 1 + 4 * s + 2 * hl + 1];
            v2f af = {y0 * y0, y1 * y1};
            su0 = wmma4(af, thB[s][0], su0);
            su1 = wmma4(af, thB[s][1], su1);
        }

        // ---- nonlinearity g(su) and weight w -> a[n,k] in LDS ----
        #pragma unroll
        for (int t = 0; t < 2; ++t) {
            v8f su = t ? su1 : su0;
            const int kk = l16 + 16 * t;
            #pragma unroll
            for (int r = 0; r < 8; ++r) {
                const int n_loc = r + 8 * hl;   // D layout: M=r (hl=0), M=r+8 (hl=1)
                float sv = su[r];               // sum_m u^2
                float c1 = 1.f / (1.f + sqrtf(1.f + sv + EPSF));
                float xn = sqrtf(sv * c1 * c1 + EPSF);
                float z  = fminf(xn, 1.f - 1e-5f);
                float g  = c1 * atanhf(z) / xn;
                int   h  = (int)ch[n_loc * ROWSZ];
                h = h < 0 ? 0 : (h > 1 ? 1 : h);
                float wv = ((n_base + n_loc) < n0) ? (h ? cw1 : cw0) : 0.f;
                aLds[wave][n_loc][kk] = wv * g;
            }
        }
        asm volatile("s_wait_dscnt 0" ::: "memory");

        // ---- GEMM2: Spart(32k x 16m) += a^T (32k x 16n) * Y (16n x 16m) ----
        #pragma unroll
        for (int s = 0; s < 4; ++s) {
            const int r0 = 4 * s + 2 * hl;      // contraction rows n
            float b0 = ch[r0 * ROWSZ + 1 + l16];
            float b1 = ch[(r0 + 1) * ROWSZ + 1 + l16];
            v2f bf  = {b0, b1};
            v2f af0 = {aLds[wave][r0][l16],      aLds[wave][r0 + 1][l16]};
            v2f af1 = {aLds[wave][r0][l16 + 16], aLds[wave][r0 + 1][l16 + 16]};
            acc0 = wmma4(af0, bf, acc0);
            acc1 = wmma4(af1, bf, acc1);
        }
        buf ^= 1;
    }

    // ---- Cross-wave reduction, applying the theta[k,m] factor ----
    #pragma unroll
    for (int t = 0; t < 2; ++t) {
        v8f acc = t ? acc1 : acc0;
        #pragma unroll
        for (int r = 0; r < 8; ++r) {
            const int k = r + 8 * hl + 16 * t;
            atomicAdd(&S_lds[k * MDIM + l16], acc[r] * theta[k * MDIM + l16]);
        }
    }
    __syncthreads();

    // ---- cumsum over k (one thread per m column) ----
    if (tid < MDIM) {
        float csum = 0.f;
        for (int k = 0; k < KDIM; ++k) {
            csum += S_lds[k * MDIM + tid];
            Scum[k * MDIM + tid] = csum;
        }
    }
    __syncthreads();

    // ---- final Poincaré map (one thread per k) ----
    if (tid < KDIM) {
        float ss = 0.f;
        #pragma unroll
        for (int m = 0; m < MDIM; ++m) {
            float v = Scum[tid * MDIM + m];
            ss += v * v;
        }
        float Sn     = sqrtf(ss + EPSF);
        float th     = tanhf(Sn);
        float factor = th / Sn;
        float xdn2   = factor * factor * ss;
        float denom  = fmaxf(1.f - xdn2, 1e-7f);
        float scale  = 2.f * factor / denom;
        float* ob = out + (size_t)b * (KDIM * MDIM) + tid * MDIM;
        #pragma unroll
        for (int m = 0; m < MDIM; ++m) ob[m] = scale * Scum[tid * MDIM + m];
    }
}

extern "C" void kernel_launch(void* const* d_in, const int* in_sizes, int n_in,
                              void* d_out, int out_size, void* d_ws, size_t ws_size,
                              hipStream_t stream) {
    const float* dgm     = (const float*)d_in[0];  // (32, 2048, 17) f32
    const float* theta   = (const float*)d_in[1];  // (32, 16) f32
    const float* class_w = (const float*)d_in[2];  // (2,) f32
    float* out = (float*)d_out;                    // (32, 512) f32
    (void)in_sizes; (void)n_in; (void)out_size; (void)d_ws; (void)ws_size;
    pmanifold_kernel<<<dim3(BATCH), dim3(256), 0, stream>>>(dgm, theta, class_w, out);
}